// DGCNN_semseg_rotate_910533067693
// MI455X (gfx1250) — compile-verified
//
#include <hip/hip_runtime.h>
#include <hip/hip_bf16.h>
#include <math.h>

// DGCNN-semseg forward for MI455X (gfx1250, wave32).
// All channel contractions on V_WMMA_F32_16X16X4_F32 (f32 keeps BN stats and
// KNN top-k indices faithful to the f32 reference).
// Round-2 tuning: 32(O) x 32(P) macro-tile per wave -- two A fragments, four
// independent WMMA accumulator chains; B fragments reused across 32 output
// rows (1.5 vmem per wmma instead of 2.5). No guards in the K loop (odd
// shapes zero-padded in workspace: C=3 -> 4, O=6 -> 32 rows).

#define B_    4
#define N_    4096
#define KNN_  20
#define EPS_  1e-5f
#define NSB_  8            // stat partial blocks (deterministic reduction)
#define WPB   4            // waves per block in WMMA kernels

typedef float v2f __attribute__((ext_vector_type(2)));
typedef float v8f __attribute__((ext_vector_type(8)));

#define WMMA_F32(a, b, c) \
    __builtin_amdgcn_wmma_f32_16x16x4_f32(false, (a), false, (b), (short)0, (c), false, false)

// ---------------------------------------------------------------------------
// WMMA GEMM: Z[b,o,p] = sum_c W[o,c] * X[b,c,p]
// Requirements: C % 4 == 0, P % (32*WPB) == 0, W has >= ceil32(O) rows of
// stride Cw (even). One wave -> 32(O) x 32(P) via four accumulators.
// ISA 7.12.2 f32 layouts:
//  A 16x4 : lanes 0-15 rows M, k=+0/+1 ; lanes 16-31 rows M, k=+2/+3
//  B 4x16 : lanes 0-15 col N, k=+0/+1 ; lanes 16-31 col N, k=+2/+3
//  D      : vgpr j -> (M = j + 8*(lane>=16), N = lane&15)
// ---------------------------------------------------------------------------
__global__ __launch_bounds__(32 * WPB)
void wmma_gemm_conv(const float* __restrict__ W, int Cw,
                    const float* __restrict__ X,
                    float* __restrict__ Z, int O, int C, int P)
{
    const int wave = threadIdx.x >> 5;
    const int lane = threadIdx.x & 31;
    const int lh   = lane & 15;
    const int khi  = (lane >> 4) << 1;                  // 0 or 2
    const int p0   = (blockIdx.x * WPB + wave) << 5;    // 32 points per wave
    const int o0   = blockIdx.y << 5;                   // 32 output rows
    const int b    = blockIdx.z;

    const float* wp0 = W + (size_t)(o0 + lh) * Cw + khi;
    const float* wp1 = wp0 + (size_t)16 * Cw;
    const float* xp  = X + (size_t)b * C * P + (size_t)khi * P + p0 + lh;

    v8f acc00 = {}, acc01 = {}, acc10 = {}, acc11 = {};
    for (int c0 = 0; c0 < C; c0 += 4) {
        v2f a0 = *reinterpret_cast<const v2f*>(wp0 + c0);
        v2f a1 = *reinterpret_cast<const v2f*>(wp1 + c0);
        const float* xc = xp + (size_t)c0 * P;
        v2f b0, b1;
        b0.x = xc[0];      b0.y = xc[P];
        b1.x = xc[16];     b1.y = xc[P + 16];
        acc00 = WMMA_F32(a0, b0, acc00);
        acc01 = WMMA_F32(a0, b1, acc01);
        acc10 = WMMA_F32(a1, b0, acc10);
        acc11 = WMMA_F32(a1, b1, acc11);
    }
    const int ob = o0 + ((lane >> 4) << 3);
    float* Zb = Z + (size_t)b * O * P + p0 + lh;
#pragma unroll
    for (int j = 0; j < 8; ++j) {
        if (ob + j < O) {
            Zb[(size_t)(ob + j) * P]      = acc00[j];
            Zb[(size_t)(ob + j) * P + 16] = acc01[j];
        }
        if (ob + 16 + j < O) {
            Zb[(size_t)(ob + 16 + j) * P]      = acc10[j];
            Zb[(size_t)(ob + 16 + j) * P + 16] = acc11[j];
        }
    }
}

// ---------------------------------------------------------------------------
// Gathered edge-conv: Z[b,o,p] = sum_c W[o,c]*X[b,c,idx[b,p]] + T[b,o,n(p)]
// W row stride wld (even), C % 4 == 0, O % 32 == 0. P = N*KNN.
// ---------------------------------------------------------------------------
__global__ __launch_bounds__(32 * WPB)
void wmma_econv_gather(const float* __restrict__ W, int wld,
                       const float* __restrict__ X,   // (B,C,N)
                       const float* __restrict__ T,   // (B,O,N)
                       const int* __restrict__ idx,   // (B,N*KNN)
                       float* __restrict__ Z,         // (B,O,N*KNN)
                       int O, int C)
{
    const int wave = threadIdx.x >> 5;
    const int lane = threadIdx.x & 31;
    const int lh   = lane & 15;
    const int khi  = (lane >> 4) << 1;
    const int P    = N_ * KNN_;
    const int p0   = (blockIdx.x * WPB + wave) << 5;
    const int o0   = blockIdx.y << 5;
    const int b    = blockIdx.z;

    const int p    = p0 + lh;
    const int col0 = idx[(size_t)b * P + p];
    const int col1 = idx[(size_t)b * P + p + 16];

    const float* wp0 = W + (size_t)(o0 + lh) * wld + khi;
    const float* wp1 = wp0 + (size_t)16 * wld;
    const float* xp  = X + (size_t)b * C * N_ + (size_t)khi * N_;

    v8f acc00 = {}, acc01 = {}, acc10 = {}, acc11 = {};
    for (int c0 = 0; c0 < C; c0 += 4) {
        v2f a0 = *reinterpret_cast<const v2f*>(wp0 + c0);
        v2f a1 = *reinterpret_cast<const v2f*>(wp1 + c0);
        const float* xc = xp + (size_t)c0 * N_;
        v2f b0, b1;
        b0.x = xc[col0];      b0.y = xc[N_ + col0];
        b1.x = xc[col1];      b1.y = xc[N_ + col1];
        acc00 = WMMA_F32(a0, b0, acc00);
        acc01 = WMMA_F32(a0, b1, acc01);
        acc10 = WMMA_F32(a1, b0, acc10);
        acc11 = WMMA_F32(a1, b1, acc11);
    }
    const int ob = o0 + ((lane >> 4) << 3);
    const int n0 = p / KNN_;
    const int n1 = (p + 16) / KNN_;
    float* Zb = Z + (size_t)b * O * P + p0 + lh;
    const float* Tb = T + (size_t)b * O * N_;
#pragma unroll
    for (int j = 0; j < 8; ++j) {
        const float* tr0 = Tb + (size_t)(ob + j) * N_;
        const float* tr1 = Tb + (size_t)(ob + 16 + j) * N_;
        Zb[(size_t)(ob + j) * P]           = acc00[j] + tr0[n0];
        Zb[(size_t)(ob + j) * P + 16]      = acc01[j] + tr0[n1];
        Zb[(size_t)(ob + 16 + j) * P]      = acc10[j] + tr1[n0];
        Zb[(size_t)(ob + 16 + j) * P + 16] = acc11[j] + tr1[n1];
    }
}

// ------------------- BatchNorm: deterministic 2-stage stats ----------------
__global__ void bn_stats(const float* __restrict__ Z, float* __restrict__ part,
                         int C, int P)
{
    const int c = blockIdx.x;
    __shared__ float s1[256], s2[256];
    float sum = 0.f, sq = 0.f;
    for (int b = 0; b < B_; ++b) {
        const float* zp = Z + ((size_t)b * C + c) * P;
        for (int i = blockIdx.y * blockDim.x + threadIdx.x; i < P;
             i += NSB_ * blockDim.x) {
            float v = zp[i];
            sum += v; sq += v * v;
        }
    }
    s1[threadIdx.x] = sum; s2[threadIdx.x] = sq;
    __syncthreads();
    for (int s = 128; s > 0; s >>= 1) {
        if (threadIdx.x < s) {
            s1[threadIdx.x] += s1[threadIdx.x + s];
            s2[threadIdx.x] += s2[threadIdx.x + s];
        }
        __syncthreads();
    }
    if (threadIdx.x == 0) {
        part[(size_t)blockIdx.y * 2 * C + c]     = s1[0];
        part[(size_t)blockIdx.y * 2 * C + C + c] = s2[0];
    }
}

__global__ void bn_finalize(const float* __restrict__ part,
                            const float* __restrict__ g,
                            const float* __restrict__ bb,
                            float* __restrict__ scsh, int C, float invCnt)
{
    int c = blockIdx.x * blockDim.x + threadIdx.x;
    if (c >= C) return;
    float sum = 0.f, sq = 0.f;
    for (int y = 0; y < NSB_; ++y) {
        sum += part[(size_t)y * 2 * C + c];
        sq  += part[(size_t)y * 2 * C + C + c];
    }
    float mean = sum * invCnt;
    float var  = sq * invCnt - mean * mean;
    float s = g[c] * rsqrtf(var + EPS_);
    scsh[c]     = s;
    scsh[C + c] = bb[c] - mean * s;
}

// act: 0 = relu, 1 = leaky-relu(0.2)
__global__ void bn_apply_act(float* __restrict__ Z, const float* __restrict__ scsh,
                             int C, int P, int act, size_t total)
{
    size_t i = (size_t)blockIdx.x * blockDim.x + threadIdx.x;
    if (i >= total) return;
    int c = (int)((i / (size_t)P) % C);
    float v = scsh[c] * Z[i] + scsh[C + c];
    Z[i] = (v >= 0.f) ? v : (act ? 0.2f * v : 0.f);
}

// fused BN + lrelu + max over k for (B,C,N,KNN) -> (B,C,N)
__global__ void bn_apply_lrelu_maxk(const float* __restrict__ Z,
                                    const float* __restrict__ scsh,
                                    float* __restrict__ out, int C)
{
    size_t i = (size_t)blockIdx.x * blockDim.x + threadIdx.x;   // B*C*N
    if (i >= (size_t)B_ * C * N_) return;
    int c = (int)((i / N_) % C);
    const float* zp = Z + i * KNN_;
    float sc = scsh[c], sh = scsh[C + c];
    float mx = -3.4e38f;
#pragma unroll
    for (int k = 0; k < KNN_; ++k) {
        float v = sc * zp[k] + sh;
        v = (v >= 0.f) ? v : 0.2f * v;
        mx = fmaxf(mx, v);
    }
    out[i] = mx;
}

__global__ void max_over_p(const float* __restrict__ Z, float* __restrict__ out,
                           int C, int P)
{   // grid (C, B)
    const int c = blockIdx.x, b = blockIdx.y;
    const float* zp = Z + ((size_t)b * C + c) * P;
    __shared__ float sm[256];
    float mx = -3.4e38f;
    for (int i = threadIdx.x; i < P; i += 256) mx = fmaxf(mx, zp[i]);
    sm[threadIdx.x] = mx;
    __syncthreads();
    for (int s = 128; s > 0; s >>= 1) {
        if (threadIdx.x < s) sm[threadIdx.x] = fmaxf(sm[threadIdx.x], sm[threadIdx.x + s]);
        __syncthreads();
    }
    if (threadIdx.x == 0) out[(size_t)b * C + c] = sm[0];
}

// ------------------------------ STN FC layers ------------------------------
// BN over batch axis only (B=4). FC bias cancels inside BN -> omitted.
__global__ void fc_bn_relu(const float* __restrict__ W, const float* __restrict__ g,
                           const float* __restrict__ bb, const float* __restrict__ in,
                           float* __restrict__ out, int O, int C)
{
    int o = blockIdx.x * blockDim.x + threadIdx.x;
    if (o >= O) return;
    float z[B_];
    float mean = 0.f;
    const float* wr = W + (size_t)o * C;
    for (int b = 0; b < B_; ++b) {
        const float* ib = in + (size_t)b * C;
        float s = 0.f;
        for (int c = 0; c < C; ++c) s += wr[c] * ib[c];
        z[b] = s; mean += s;
    }
    mean *= (1.f / B_);
    float var = 0.f;
    for (int b = 0; b < B_; ++b) { float d = z[b] - mean; var += d * d; }
    var *= (1.f / B_);
    float sc = g[o] * rsqrtf(var + EPS_);
    for (int b = 0; b < B_; ++b) {
        float v = sc * (z[b] - mean) + bb[o];
        out[(size_t)b * O + o] = fmaxf(v, 0.f);
    }
}

__global__ void fc3_trans(const float* __restrict__ W, const float* __restrict__ bias,
                          const float* __restrict__ in, float* __restrict__ tws,
                          float* __restrict__ tout)
{
    int t = threadIdx.x;
    if (t >= B_ * 9) return;
    int b = t / 9, o = t % 9;
    float s = bias[o] + ((o == 0 || o == 4 || o == 8) ? 1.f : 0.f);
    const float* ib = in + b * 256;
    const float* wr = W + o * 256;
    for (int c = 0; c < 256; ++c) s += wr[c] * ib[c];
    tws[t]  = s;
    tout[t] = s;
}

// xt4[b,d,n] (d<3) = sum_c x[b,c,n]*tr[b,c,d]; channel 3 zero-padded
__global__ void apply_trans4(const float* __restrict__ x, const float* __restrict__ tr,
                             float* __restrict__ xt4)
{
    int i = blockIdx.x * blockDim.x + threadIdx.x;
    if (i >= B_ * 4 * N_) return;
    int n = i % N_, d = (i / N_) % 4, b = i / (4 * N_);
    if (d == 3) { xt4[i] = 0.f; return; }
    const float* xb = x + (size_t)b * 3 * N_;
    const float* tb = tr + b * 9;
    xt4[i] = xb[n] * tb[d] + xb[N_ + n] * tb[3 + d] + xb[2 * N_ + n] * tb[6 + d];
}

// x (B,3,N) -> x4 (B,4,N) zero-padded
__global__ void pad_x4(const float* __restrict__ x, float* __restrict__ x4)
{
    int i = blockIdx.x * blockDim.x + threadIdx.x;
    if (i >= B_ * 4 * N_) return;
    int n = i % N_, c = (i / N_) % 4, b = i / (4 * N_);
    x4[i] = (c < 3) ? x[((size_t)b * 3 + c) * N_ + n] : 0.f;
}

// zero-padded weight copy: src (O x C, row stride srcStride) -> dst (Opad x Cpad)
__global__ void pad_w(const float* __restrict__ src, int srcStride, int O, int C,
                      float* __restrict__ dst, int Opad, int Cpad)
{
    int i = blockIdx.x * blockDim.x + threadIdx.x;
    if (i >= Opad * Cpad) return;
    int o = i / Cpad, c = i % Cpad;
    dst[i] = (o < O && c < C) ? src[(size_t)o * srcStride + c] : 0.f;
}

// center-weight fold: weff[o,c] = w[o, C+c] - w[o, c]  (w is O x 2C), Cpad out
__global__ void prep_weff(const float* __restrict__ w, float* __restrict__ weff,
                          int O, int C, int Cpad)
{
    int i = blockIdx.x * blockDim.x + threadIdx.x;
    if (i >= O * Cpad) return;
    int o = i / Cpad, c = i % Cpad;
    weff[i] = (c < C) ? (w[(size_t)o * 2 * C + C + c] - w[(size_t)o * 2 * C + c]) : 0.f;
}

// ------------------------------- KNN (fused) -------------------------------
__global__ void compute_xx(const float* __restrict__ X, float* __restrict__ xx, int C)
{
    int i = blockIdx.x * blockDim.x + threadIdx.x;   // B*N
    if (i >= B_ * N_) return;
    int b = i / N_, n = i % N_;
    const float* xb = X + (size_t)b * C * N_ + n;
    float s = 0.f;
    for (int c = 0; c < C; ++c) { float v = xb[(size_t)c * N_]; s += v * v; }
    xx[i] = s;
}

// d = 2*inner - xx[n] - xx[m]; top-KNN by insertion (stable ties == jax top_k)
// X has CROWS rows allocated (row stride N_), first CC used.
template <int CC, int CROWS>
__global__ __launch_bounds__(128)
void knn_topk(const float* __restrict__ X, const float* __restrict__ xx,
              int* __restrict__ idx)
{
    const int b = blockIdx.z;
    const int n = blockIdx.x * 128 + threadIdx.x;
    const float* Xb = X + (size_t)b * CROWS * N_;
    float q[CC];
#pragma unroll
    for (int c = 0; c < CC; ++c) q[c] = Xb[(size_t)c * N_ + n];
    const float qq = xx[b * N_ + n];
    float vals[KNN_]; int ids[KNN_];
#pragma unroll
    for (int k = 0; k < KNN_; ++k) { vals[k] = -3.4e38f; ids[k] = 0; }
    __shared__ float tile[CC * 32 + 32];
    for (int m0 = 0; m0 < N_; m0 += 32) {
        __syncthreads();
        for (int t = threadIdx.x; t < CC * 32; t += 128)
            tile[t] = Xb[(size_t)(t / 32) * N_ + m0 + (t & 31)];
        if (threadIdx.x < 32) tile[CC * 32 + threadIdx.x] = xx[b * N_ + m0 + threadIdx.x];
        __syncthreads();
        for (int j = 0; j < 32; ++j) {
            float inner = 0.f;
#pragma unroll
            for (int c = 0; c < CC; ++c) inner += q[c] * tile[c * 32 + j];
            float d = 2.f * inner - qq - tile[CC * 32 + j];
            if (d > vals[KNN_ - 1]) {
                float cd = d; int cm = m0 + j;
#pragma unroll
                for (int k = 0; k < KNN_; ++k) {
                    if (cd > vals[k]) {
                        float tv = vals[k]; vals[k] = cd; cd = tv;
                        int ti = ids[k]; ids[k] = cm; cm = ti;
                    }
                }
            }
        }
    }
    int* op = idx + ((size_t)b * N_ + n) * KNN_;
#pragma unroll
    for (int k = 0; k < KNN_; ++k) op[k] = ids[k];
}

// --------------------------------- concats ---------------------------------
__global__ void concat3(const float* __restrict__ x1, const float* __restrict__ x2,
                        const float* __restrict__ x3, float* __restrict__ out)
{
    size_t i = (size_t)blockIdx.x * blockDim.x + threadIdx.x;   // B*192*N
    if (i >= (size_t)B_ * 192 * N_) return;
    int n = (int)(i % N_);
    int c = (int)((i / N_) % 192);
    int b = (int)(i / ((size_t)192 * N_));
    const float* s = (c < 64) ? x1 : (c < 128 ? x2 : x3);
    int cc = (c < 64) ? c : (c < 128 ? c - 64 : c - 128);
    out[i] = s[((size_t)b * 64 + cc) * N_ + n];
}

__global__ void concat_g(const float* __restrict__ g, const float* __restrict__ x1,
                         const float* __restrict__ x2, const float* __restrict__ x3,
                         float* __restrict__ out)
{
    size_t i = (size_t)blockIdx.x * blockDim.x + threadIdx.x;   // B*1216*N
    if (i >= (size_t)B_ * 1216 * N_) return;
    int n = (int)(i % N_);
    int c = (int)((i / N_) % 1216);
    int b = (int)(i / ((size_t)1216 * N_));
    float v;
    if (c < 1024) {
        v = g[(size_t)b * 1024 + c];
    } else {
        int c2 = c - 1024;
        const float* s = (c2 < 64) ? x1 : (c2 < 128 ? x2 : x3);
        int cc = (c2 < 64) ? c2 : (c2 < 128 ? c2 - 64 : c2 - 128);
        v = s[((size_t)b * 64 + cc) * N_ + n];
    }
    out[i] = v;
}

// ===========================================================================
extern "C" void kernel_launch(void* const* d_in, const int* in_sizes, int n_in,
                              void* d_out, int out_size, void* d_ws, size_t ws_size,
                              hipStream_t stream)
{
    (void)in_sizes; (void)n_in; (void)out_size; (void)ws_size;
    // d_in order: top-level dict insertion order {x, aux, params}; params is a
    // jax pytree -> leaves in sorted-key order, 'stn' sub-dict expanded in place.
    auto f = [&](int i) { return (const float*)d_in[i]; };
    const float* x = f(0);
    const float *bt1 = f(2),  *bt2 = f(3),  *bt3 = f(4),  *bt4 = f(5);
    const float *bt5 = f(6),  *bt6 = f(7),  *bt7 = f(8),  *bt8 = f(9);
    const float *gm1 = f(10), *gm2 = f(11), *gm3 = f(12), *gm4 = f(13);
    const float *gm5 = f(14), *gm6 = f(15), *gm7 = f(16), *gm8 = f(17);
    const float *s_bb1 = f(18), *s_bb2 = f(19), *s_bb3 = f(20), *s_bb4 = f(21), *s_bb5 = f(22);
    const float *s_bg1 = f(23), *s_bg2 = f(24), *s_bg3 = f(25), *s_bg4 = f(26), *s_bg5 = f(27);
    // 28..30: cb1..cb3 (conv bias before BN: provably no-op, skipped)
    const float *s_cw1 = f(31), *s_cw2 = f(32), *s_cw3 = f(33);
    // 34,35: fb1,fb2 cancel in BN
    const float *s_fb3 = f(36);
    const float *s_fw1 = f(37), *s_fw2 = f(38), *s_fw3 = f(39);
    const float *w1 = f(40), *w2 = f(41), *w3 = f(42), *w4 = f(43), *w5 = f(44);
    const float *w6 = f(45), *w7 = f(46), *w8 = f(47), *w9 = f(48);

    // ---- workspace layout (floats, 256B aligned blocks) ----
    float* wsf = (float*)d_ws;
    size_t off = 0;
    auto alloc = [&](size_t nf) {
        float* p = wsf + off;
        off += (nf + 63) & ~(size_t)63;
        return p;
    };
    float* x4in   = alloc((size_t)B_ * 4 * N_);          // raw x, C padded to 4
    float* xt4    = alloc((size_t)B_ * 4 * N_);          // transformed x, padded
    float* h1     = alloc((size_t)B_ * 64 * N_);
    float* h2     = alloc((size_t)B_ * 128 * N_);
    float* hbig   = alloc((size_t)B_ * 1024 * N_);       // stn z3 / pconv6 z / z8
    float* pool   = alloc((size_t)B_ * 1024);            // stn pool, later g-max
    float* fc1o   = alloc((size_t)B_ * 512);
    float* fc2o   = alloc((size_t)B_ * 256);
    float* trb    = alloc(64);
    float* xx     = alloc((size_t)B_ * N_);
    int*   knn    = (int*)alloc((size_t)B_ * N_ * KNN_);
    float* teff   = alloc((size_t)B_ * 64 * N_);
    float* weffp  = alloc(64 * 64);
    float* cw1p   = alloc(64 * 4);
    float* w1nbp  = alloc(64 * 4);
    float* w9p    = alloc(32 * 256);
    float* part   = alloc((size_t)NSB_ * 2 * 1024);
    float* scsh   = alloc(2 * 1024);
    float* xs1    = alloc((size_t)B_ * 64 * N_);
    float* xs2    = alloc((size_t)B_ * 64 * N_);
    float* xs3    = alloc((size_t)B_ * 64 * N_);
    float* cat192 = alloc((size_t)B_ * 192 * N_);
    float* zA     = alloc((size_t)B_ * 64 * N_ * KNN_);  // reused as cat1216
    float* zB     = alloc((size_t)B_ * 64 * N_ * KNN_);  // reused as z7

    const int PNK = N_ * KNN_;

    auto gemm = [&](const float* W, int Cw, const float* X, float* Z,
                    int O, int C, int P) {
        dim3 g(P / (32 * WPB), (O + 31) / 32, B_);
        wmma_gemm_conv<<<g, 32 * WPB, 0, stream>>>(W, Cw, X, Z, O, C, P);
    };
    auto egather = [&](const float* W, int wld, const float* X, float* Z, int C) {
        dim3 g(PNK / (32 * WPB), 2, B_);
        wmma_econv_gather<<<g, 32 * WPB, 0, stream>>>(W, wld, X, teff, knn, Z, 64, C);
    };
    auto stats = [&](float* z, int C, int P, const float* g, const float* bb, float cnt) {
        dim3 gs(C, NSB_);
        bn_stats<<<gs, 256, 0, stream>>>(z, part, C, P);
        bn_finalize<<<(C + 127) / 128, 128, 0, stream>>>(part, g, bb, scsh, C, 1.f / cnt);
    };
    auto apply = [&](float* z, int C, int P, int act) {
        size_t total = (size_t)B_ * C * P;
        bn_apply_act<<<(unsigned)((total + 255) / 256), 256, 0, stream>>>(z, scsh, C, P, act, total);
    };
    auto maxk = [&](const float* z, float* out) {
        size_t total = (size_t)B_ * 64 * N_;
        bn_apply_lrelu_maxk<<<(unsigned)((total + 255) / 256), 256, 0, stream>>>(z, scsh, out, 64);
    };

    // ------------------------- padded operand prep -----------------------
    pad_x4<<<(B_ * 4 * N_ + 255) / 256, 256, 0, stream>>>(x, x4in);
    pad_w<<<1, 256, 0, stream>>>(s_cw1, 3, 64, 3, cw1p, 64, 4);    // 256 elems
    pad_w<<<1, 256, 0, stream>>>(w1, 6, 64, 3, w1nbp, 64, 4);      // w1[:, :3]
    pad_w<<<32, 256, 0, stream>>>(w9, 256, 6, 256, w9p, 32, 256);  // 8192 elems

    // ------------------------------- STN --------------------------------
    gemm(cw1p, 4, x4in, h1, 64, 4, N_);
    stats(h1, 64, N_, s_bg1, s_bb1, (float)B_ * N_);
    apply(h1, 64, N_, 0);
    gemm(s_cw2, 64, h1, h2, 128, 64, N_);
    stats(h2, 128, N_, s_bg2, s_bb2, (float)B_ * N_);
    apply(h2, 128, N_, 0);
    gemm(s_cw3, 128, h2, hbig, 1024, 128, N_);
    stats(hbig, 1024, N_, s_bg3, s_bb3, (float)B_ * N_);
    apply(hbig, 1024, N_, 0);
    max_over_p<<<dim3(1024, B_), 256, 0, stream>>>(hbig, pool, 1024, N_);
    fc_bn_relu<<<4, 128, 0, stream>>>(s_fw1, s_bg4, s_bb4, pool, fc1o, 512, 1024);
    fc_bn_relu<<<2, 128, 0, stream>>>(s_fw2, s_bg5, s_bb5, fc1o, fc2o, 256, 512);
    fc3_trans<<<1, 64, 0, stream>>>(s_fw3, s_fb3, fc2o, trb, (float*)d_out + (size_t)B_ * 6 * N_);
    apply_trans4<<<(B_ * 4 * N_ + 255) / 256, 256, 0, stream>>>(x, trb, xt4);

    // ------------------------------ Stage 1 ------------------------------
    compute_xx<<<(B_ * N_ + 255) / 256, 256, 0, stream>>>(xt4, xx, 3);
    { dim3 g(N_ / 128, 1, B_); knn_topk<3, 4><<<g, 128, 0, stream>>>(xt4, xx, knn); }
    prep_weff<<<1, 256, 0, stream>>>(w1, weffp, 64, 3, 4);
    gemm(weffp, 4, xt4, teff, 64, 4, N_);
    egather(w1nbp, 4, xt4, zA, 4);
    stats(zA, 64, PNK, gm1, bt1, (float)B_ * PNK);
    apply(zA, 64, PNK, 1);
    gemm(w2, 64, zA, zB, 64, 64, PNK);
    stats(zB, 64, PNK, gm2, bt2, (float)B_ * PNK);
    maxk(zB, xs1);

    // ------------------------------ Stage 2 ------------------------------
    compute_xx<<<(B_ * N_ + 255) / 256, 256, 0, stream>>>(xs1, xx, 64);
    { dim3 g(N_ / 128, 1, B_); knn_topk<64, 64><<<g, 128, 0, stream>>>(xs1, xx, knn); }
    prep_weff<<<16, 256, 0, stream>>>(w3, weffp, 64, 64, 64);
    gemm(weffp, 64, xs1, teff, 64, 64, N_);
    egather(w3, 128, xs1, zA, 64);
    stats(zA, 64, PNK, gm3, bt3, (float)B_ * PNK);
    apply(zA, 64, PNK, 1);
    gemm(w4, 64, zA, zB, 64, 64, PNK);
    stats(zB, 64, PNK, gm4, bt4, (float)B_ * PNK);
    maxk(zB, xs2);

    // ------------------------------ Stage 3 ------------------------------
    compute_xx<<<(B_ * N_ + 255) / 256, 256, 0, stream>>>(xs2, xx, 64);
    { dim3 g(N_ / 128, 1, B_); knn_topk<64, 64><<<g, 128, 0, stream>>>(xs2, xx, knn); }
    prep_weff<<<16, 256, 0, stream>>>(w5, weffp, 64, 64, 64);
    gemm(weffp, 64, xs2, teff, 64, 64, N_);
    egather(w5, 128, xs2, zA, 64);
    stats(zA, 64, PNK, gm5, bt5, (float)B_ * PNK);
    maxk(zA, xs3);

    // ----------------------------- Global head ---------------------------
    {
        size_t t = (size_t)B_ * 192 * N_;
        concat3<<<(unsigned)((t + 255) / 256), 256, 0, stream>>>(xs1, xs2, xs3, cat192);
    }
    gemm(w6, 192, cat192, hbig, 1024, 192, N_);
    stats(hbig, 1024, N_, gm6, bt6, (float)B_ * N_);
    apply(hbig, 1024, N_, 1);
    max_over_p<<<dim3(1024, B_), 256, 0, stream>>>(hbig, pool, 1024, N_);
    {
        size_t t = (size_t)B_ * 1216 * N_;
        concat_g<<<(unsigned)((t + 255) / 256), 256, 0, stream>>>(pool, xs1, xs2, xs3, zA);
    }
    gemm(w7, 1216, zA, zB, 512, 1216, N_);
    stats(zB, 512, N_, gm7, bt7, (float)B_ * N_);
    apply(zB, 512, N_, 1);
    gemm(w8, 512, zB, hbig, 256, 512, N_);
    stats(hbig, 256, N_, gm8, bt8, (float)B_ * N_);
    apply(hbig, 256, N_, 1);
    gemm(w9p, 256, hbig, (float*)d_out, 6, 256, N_);   // (B,6,N) -> d_out head
}